// CRFModel_14851996909756
// MI455X (gfx1250) — compile-verified
//
#include <hip/hip_runtime.h>

// ---------------------------------------------------------------------------
// CRF logZ on gfx1250 (MI455X).
//   K=64 tags, V=50257 vocab, D=512, BT=256 sentences, T=128 steps.
// Phase 1: Bt[v][k] = ThetaB[k,:] . E[v,:]  via V_WMMA_F32_16X16X4_F32
//          (vocab-major so the per-word gather is 64 contiguous floats).
// Phase 2: forward recursion alpha' = logsumexp_s(alpha_s + A[s,j]) + emis.
// ---------------------------------------------------------------------------

#define KTAGS 64
#define VOCAB 50257
#define DIM   512
#define BTCH  256
#define TLEN  128
#define BOS_T 62
#define EOS_T 63
#define NEGV  (-1e30f)

#define VTILE 128                                        // vocab words per block
#define VPAD  (((VOCAB + VTILE - 1) / VTILE) * VTILE)    // 50304
#define DCHUNK 128                                       // D staged per LDS chunk
#define LDS_STRIDE 132                                   // 132 % 64 == 4 -> no bank conflicts

typedef float v2f __attribute__((ext_vector_type(2)));
typedef float v8f __attribute__((ext_vector_type(8)));

// ---------------------------------------------------------------------------
// Kernel 1: emission GEMM. One wave = 16 vocab words x all 64 tags.
// A (ThetaB tile) from LDS, B (E rows) straight from global (E read once,
// HBM-bound anyway). f32 WMMA keeps reference precision.
// ---------------------------------------------------------------------------
__global__ __launch_bounds__(256) void emit_gemm(const float* __restrict__ ThetaB,
                                                 const float* __restrict__ E,
                                                 float* __restrict__ Bt) {
  __shared__ float chunk[KTAGS * LDS_STRIDE];            // 64 x 128 (+pad) f32 ~ 33 KB

  const int tid  = threadIdx.x;
  const int wave = tid >> 5;          // 0..7
  const int lane = tid & 31;
  const int g    = lane >> 4;         // lane group: K-pair selector
  const int ln   = lane & 15;         // M (A) / N (B,C) index within tile

  const int n0 = blockIdx.x * VTILE + wave * 16;         // vocab tile base
  int erow = n0 + ln;
  if (erow >= VOCAB) erow = VOCAB - 1;                   // clamp: no OOB read of E
  const float* __restrict__ Ep = E + (size_t)erow * DIM;

  v8f c0 = {}, c1 = {}, c2 = {}, c3 = {};

  for (int dc = 0; dc < DIM; dc += DCHUNK) {
    __syncthreads();                                     // previous chunk fully consumed
    for (int i = tid; i < KTAGS * DCHUNK; i += 256) {
      const int r = i >> 7, cidx = i & (DCHUNK - 1);
      chunk[r * LDS_STRIDE + cidx] = ThetaB[r * DIM + dc + cidx];
    }
    __syncthreads();

    #pragma unroll 4
    for (int dd = 0; dd < DCHUNK; dd += 4) {
      const int kk = dd + 2 * g;                         // ISA 16x4 f32 frag: K = 2*g + vgpr
      const v2f b  = *(const v2f*)(Ep + dc + kk);        // B[kk..kk+1][n] = E[word][dc+kk..]
      const v2f a0 = *(const v2f*)(chunk + ( 0 + ln) * LDS_STRIDE + kk);
      const v2f a1 = *(const v2f*)(chunk + (16 + ln) * LDS_STRIDE + kk);
      const v2f a2 = *(const v2f*)(chunk + (32 + ln) * LDS_STRIDE + kk);
      const v2f a3 = *(const v2f*)(chunk + (48 + ln) * LDS_STRIDE + kk);
      c0 = __builtin_amdgcn_wmma_f32_16x16x4_f32(false, a0, false, b, (short)0, c0, false, false);
      c1 = __builtin_amdgcn_wmma_f32_16x16x4_f32(false, a1, false, b, (short)0, c1, false, false);
      c2 = __builtin_amdgcn_wmma_f32_16x16x4_f32(false, a2, false, b, (short)0, c2, false, false);
      c3 = __builtin_amdgcn_wmma_f32_16x16x4_f32(false, a3, false, b, (short)0, c3, false, false);
    }
  }

  // C layout: VGPR r, lane -> element (M = r + 8*g, N = ln). Store vocab-major,
  // masking emissions for BOS/EOS tags (62, 63) to -1e30.
  float* __restrict__ orow = Bt + (size_t)(n0 + ln) * KTAGS;
  #pragma unroll
  for (int r = 0; r < 8; ++r) {
    orow[ 0 + 8 * g + r] = c0[r];
    orow[16 + 8 * g + r] = c1[r];
    orow[32 + 8 * g + r] = c2[r];
    const int k3 = 48 + 8 * g + r;
    orow[k3] = (k3 >= BOS_T) ? NEGV : c3[r];
  }
}

// ---------------------------------------------------------------------------
// Kernel 2: forward recursion. 256 threads = 4 sentences x 64 states.
// ---------------------------------------------------------------------------
__global__ __launch_bounds__(256) void crf_forward(const int* __restrict__ words,
                                                   const float* __restrict__ WA,
                                                   const float* __restrict__ Bt,
                                                   float* __restrict__ out) {
  __shared__ float Atr[KTAGS * KTAGS];   // A[s][j], BOS column masked (16 KB)
  __shared__ float alpha[4][KTAGS];
  __shared__ int   wbuf[4][TLEN];

  const int tid  = threadIdx.x;
  const int sent = tid >> 6;             // local sentence 0..3
  const int j    = tid & 63;             // state
  const int b    = blockIdx.x * 4 + sent;

  for (int i = tid; i < KTAGS * KTAGS; i += 256)
    Atr[i] = ((i & 63) == BOS_T) ? NEGV : WA[i];         // no transitions into BOS
  for (int i = tid; i < 4 * TLEN; i += 256)
    wbuf[i >> 7][i & (TLEN - 1)] =
        words[((size_t)blockIdx.x * 4 + (i >> 7)) * TLEN + (i & (TLEN - 1))];
  alpha[sent][j] = (j == BOS_T) ? 0.0f : NEGV;           // start in BOS
  __syncthreads();

  for (int t = 0; t < TLEN; ++t) {
    const float e = Bt[(size_t)wbuf[sent][t] * KTAGS + j];   // 64-contig gather
    const float* __restrict__ col = Atr + j;

    float m = alpha[sent][0] + col[0];
    #pragma unroll 8
    for (int s = 1; s < KTAGS; ++s)
      m = fmaxf(m, alpha[sent][s] + col[s * KTAGS]);
    float sum = 0.0f;
    #pragma unroll 8
    for (int s = 0; s < KTAGS; ++s)
      sum += __expf(alpha[sent][s] + col[s * KTAGS] - m);
    const float na = m + __logf(sum) + e;

    __syncthreads();
    alpha[sent][j] = na;
    __syncthreads();
  }

  // final transition into EOS + logsumexp over states
  const float r = alpha[sent][j] + Atr[j * KTAGS + EOS_T];
  __syncthreads();
  alpha[sent][j] = r;
  __syncthreads();
  if (j == 0) {
    float m = alpha[sent][0];
    for (int s = 1; s < KTAGS; ++s) m = fmaxf(m, alpha[sent][s]);
    float sum = 0.0f;
    for (int s = 0; s < KTAGS; ++s) sum += __expf(alpha[sent][s] - m);
    out[b] = m + __logf(sum);
  }
}

// ---------------------------------------------------------------------------
extern "C" void kernel_launch(void* const* d_in, const int* in_sizes, int n_in,
                              void* d_out, int out_size, void* d_ws, size_t ws_size,
                              hipStream_t stream) {
  (void)in_sizes; (void)n_in; (void)out_size; (void)ws_size;
  const int*   words  = (const int*)d_in[0];
  const float* WA     = (const float*)d_in[1];
  const float* ThetaB = (const float*)d_in[2];
  const float* E      = (const float*)d_in[3];
  float* Bt  = (float*)d_ws;              // needs VPAD*64*4 = ~12.9 MB scratch
  float* out = (float*)d_out;

  emit_gemm<<<VPAD / VTILE, 256, 0, stream>>>(ThetaB, E, Bt);
  crf_forward<<<BTCH / 4, 256, 0, stream>>>(words, WA, Bt, out);
}